// Generator_34514357190755
// MI455X (gfx1250) — compile-verified
//
#include <hip/hip_runtime.h>
#include <hip/hip_bf16.h>
#include <math.h>
#include <stdint.h>

#define B_ 16
#define L_ 400
#define H_ 400
#define V_ 18000
#define S_ 30
#define T_ 10
#define G_ 3

// WMMA GEMM blocking: 5 waves/block, 5 16-col tiles/wave -> 400 cols per block.
// 3H = 1200 = 3 blocks, V = 18000 = 45 blocks (both exact; no predication, EXEC stays all-1s).
#define WV 5
#define NTW 5
#define BLK_COLS (WV * NTW * 16)
#define APAD 402  // padded LDS row stride (floats): even (b64-aligned), 18*m mod 64 distinct -> conflict-free

typedef float v2f __attribute__((ext_vector_type(2)));
typedef float v8f __attribute__((ext_vector_type(8)));

__device__ __forceinline__ float sigm(float x) { return 1.0f / (1.0f + __expf(-x)); }

// ---------- block reductions (wave32-aware) ----------
__device__ __forceinline__ float block_max(float v, float* red) {
  for (int o = 16; o > 0; o >>= 1) v = fmaxf(v, __shfl_down(v, o, 32));
  int w = threadIdx.x >> 5, l = threadIdx.x & 31;
  __syncthreads();
  if (l == 0) red[w] = v;
  __syncthreads();
  if (threadIdx.x == 0) {
    int nw = (blockDim.x + 31) >> 5;
    float m = red[0];
    for (int i = 1; i < nw; ++i) m = fmaxf(m, red[i]);
    red[0] = m;
  }
  __syncthreads();
  return red[0];
}

__device__ __forceinline__ float block_sum(float v, float* red) {
  for (int o = 16; o > 0; o >>= 1) v += __shfl_down(v, o, 32);
  int w = threadIdx.x >> 5, l = threadIdx.x & 31;
  __syncthreads();
  if (l == 0) red[w] = v;
  __syncthreads();
  if (threadIdx.x == 0) {
    int nw = (blockDim.x + 31) >> 5;
    float m = red[0];
    for (int i = 1; i < nw; ++i) m += red[i];
    red[0] = m;
  }
  __syncthreads();
  return red[0];
}

// ---------- WMMA GEMM with LDS-staged A:  G[s,b,n] = sum_k X[s,b,k]*W[n,k] (+ bias[n]) ----------
// X: [S,B,H], W: [N,H], Gout: [S,B,N]. Block: 5 waves, covers 400 columns of one slot s.
// A (16 x 400 f32, 25.6KB) is staged once per block into LDS with async global->LDS DMA
// (GLOBAL_LOAD_ASYNC_TO_LDS_B64, ASYNCcnt-tracked), then shared by all 25 output tiles.
// The LDS-offset operand is derived from &shA[...] itself (ptrtoint): this CAPTURES shA so
// the compiler must treat the asm as writing it, keeping the ds_load reads live.
// Per wave: K-loop of 100 iterations x 5 V_WMMA_F32_16X16X4_F32 reusing one A fragment.
__global__ __launch_bounds__(WV * 32) void wmma_gemm_lds_kernel(
    const float* __restrict__ X, const float* __restrict__ W,
    const float* __restrict__ bias, float* __restrict__ Gout, int N) {
  __shared__ float shA[16 * APAD];
  const int tid  = threadIdx.x;
  const int wave = tid >> 5;
  const int lane = tid & 31;
  const int s    = blockIdx.y;
  const float* Xs = X + (size_t)s * B_ * H_;

  // ---- stage A[16 x H] -> LDS via async copy (8B per lane-op; 3200/160 = 20 ops/lane) ----
  for (int i = tid; i < 16 * (H_ / 2); i += WV * 32) {
    const int r = i / (H_ / 2);
    const int c = (i - r * (H_ / 2)) * 2;
    // low 32 bits of the generic LDS address == wave-relative LDS byte offset;
    // ptrtoint also forces shA to escape into the asm (memory clobber now applies).
    const unsigned ldsoff = (unsigned)(uintptr_t)&shA[r * APAD + c];
    const unsigned long long gaddr =
        (unsigned long long)(uintptr_t)(Xs + (size_t)r * H_ + c);
    asm volatile("global_load_async_to_lds_b64 %0, %1, off"
                 :: "v"(ldsoff), "v"(gaddr) : "memory");
  }
  asm volatile("s_wait_asynccnt 0" ::: "memory");
  __syncthreads();

  const int m   = lane & 15;  // batch row (A), column-within-tile (B/D)
  const int kh  = lane >> 4;  // K half-select (0 or 1)
  const int nw0 = blockIdx.x * BLK_COLS + wave * (NTW * 16);

  v8f acc[NTW];
#pragma unroll
  for (int tt = 0; tt < NTW; ++tt) {
    const float bval = bias ? bias[nw0 + tt * 16 + m] : 0.0f;
#pragma unroll
    for (int r = 0; r < 8; ++r) acc[tt][r] = bval;
  }

  const float* wbase = W + (size_t)(nw0 + m) * H_;
  for (int k0 = 0; k0 < H_; k0 += 4) {
    const int k = k0 + 2 * kh;
    v2f a;
    a[0] = shA[m * APAD + k];
    a[1] = shA[m * APAD + k + 1];
#pragma unroll
    for (int tt = 0; tt < NTW; ++tt) {
      const float* wr = wbase + (size_t)tt * 16 * H_;
      v2f b;
      b[0] = wr[k];
      b[1] = wr[k + 1];
      acc[tt] = __builtin_amdgcn_wmma_f32_16x16x4_f32(
          /*neg_a=*/false, a, /*neg_b=*/false, b,
          /*c_mod=*/(short)0, acc[tt], /*reuse_a=*/false, /*reuse_b=*/false);
    }
  }

  const int mb = kh * 8;
#pragma unroll
  for (int tt = 0; tt < NTW; ++tt) {
    const int n = nw0 + tt * 16 + m;
    float* gp = Gout + (size_t)s * B_ * N + n;
#pragma unroll
    for (int r = 0; r < 8; ++r) gp[(size_t)(mb + r) * N] = acc[tt][r];
  }
}

// ---------- GRU elementwise combine ----------
__global__ void gru_elem_kernel(const float* __restrict__ gi, const float* __restrict__ gh,
                                const float* __restrict__ hPrev, float* __restrict__ hCur) {
  int idx = blockIdx.x * blockDim.x + threadIdx.x;
  if (idx >= S_ * B_ * H_) return;
  int j = idx % H_;
  int sb = idx / H_;
  const float* giP = gi + (size_t)sb * 3 * H_;
  const float* ghP = gh + (size_t)sb * 3 * H_;
  float r = sigm(giP[j] + ghP[j]);
  float z = sigm(giP[H_ + j] + ghP[H_ + j]);
  float n = tanhf(giP[2 * H_ + j] + r * ghP[2 * H_ + j]);
  hCur[idx] = (1.0f - z) * n + z * hPrev[idx];
}

// ---------- attention + softmax + context + pgen + (t==0) gate ----------
__global__ __launch_bounds__(256) void attn_kernel(
    const float* __restrict__ hNew, const float* __restrict__ decIn,
    const float* __restrict__ enc, const int* __restrict__ lens,
    const float* __restrict__ Wg, const float* __restrict__ bg,
    const float* __restrict__ Wr, const float* __restrict__ br,
    float* __restrict__ prob, float* __restrict__ pgen,
    float* __restrict__ gates, int t) {
  __shared__ float sh_h[H_];
  __shared__ float sh_sc[L_];
  __shared__ float sh_ctx[H_];
  __shared__ float red[32];
  const int b = blockIdx.x, s = blockIdx.y, tid = threadIdx.x, nthr = blockDim.x;

  const float* hv = hNew + ((size_t)s * B_ + b) * H_;
  const float* dv = decIn + ((size_t)s * B_ + b) * H_;
  const float* eb = enc + (size_t)b * L_ * H_;

  for (int j = tid; j < H_; j += nthr) sh_h[j] = hv[j];
  __syncthreads();

  const int len = lens[b];
  for (int l = tid; l < L_; l += nthr) {
    const float* er = eb + (size_t)l * H_;
    float d = 0.0f;
    for (int k = 0; k < H_; ++k) d += sh_h[k] * er[k];
    sh_sc[l] = (l < len) ? d : -3.0e38f;
  }
  __syncthreads();

  float mx = -3.0e38f;
  for (int l = tid; l < L_; l += nthr) mx = fmaxf(mx, sh_sc[l]);
  mx = block_max(mx, red);

  float sm = 0.0f;
  for (int l = tid; l < L_; l += nthr) { float e = __expf(sh_sc[l] - mx); sh_sc[l] = e; sm += e; }
  __syncthreads();
  sm = block_sum(sm, red);
  const float inv = 1.0f / sm;
  for (int l = tid; l < L_; l += nthr) {
    float p = sh_sc[l] * inv;
    sh_sc[l] = p;
    prob[((size_t)s * B_ + b) * L_ + l] = p;
  }
  __syncthreads();

  for (int j = tid; j < H_; j += nthr) {
    float c = 0.0f;
    for (int l = 0; l < L_; ++l) c += sh_sc[l] * eb[(size_t)l * H_ + j];
    sh_ctx[j] = c;
  }
  __syncthreads();

  // pgen = sigmoid([h, ctx, dec_in] . W_ratio + b)
  float p = 0.0f;
  for (int j = tid; j < H_; j += nthr)
    p += sh_h[j] * Wr[j] + sh_ctx[j] * Wr[H_ + j] + dv[j] * Wr[2 * H_ + j];
  p = block_sum(p, red);
  if (tid == 0) pgen[(size_t)s * B_ + b] = sigm(p + br[0]);

  if (t == 0) {
    for (int g = 0; g < G_; ++g) {
      float gv = 0.0f;
      for (int j = tid; j < H_; j += nthr) gv += sh_ctx[j] * Wg[g * H_ + j];
      gv = block_sum(gv, red);
      if (tid == 0) gates[((size_t)s * B_ + b) * G_ + g] = gv + bg[g];
    }
  }
}

// ---------- vocab softmax + pgen*p_vocab write ----------
__global__ __launch_bounds__(256) void vocab_softmax_kernel(
    const float* __restrict__ logits, const float* __restrict__ pgen,
    float* __restrict__ out, int t) {
  __shared__ float red[32];
  const int b = blockIdx.x, s = blockIdx.y, tid = threadIdx.x, nthr = blockDim.x;
  const float* lg = logits + ((size_t)s * B_ + b) * V_;

  float mx = -3.0e38f;
  for (int v = tid; v < V_; v += nthr) mx = fmaxf(mx, lg[v]);
  mx = block_max(mx, red);

  float sm = 0.0f;
  for (int v = tid; v < V_; v += nthr) sm += __expf(lg[v] - mx);
  sm = block_sum(sm, red);

  const float scale = pgen[(size_t)s * B_ + b] / sm;
  float* op = out + (((size_t)s * B_ + b) * T_ + t) * (size_t)V_;
  for (int v = tid; v < V_; v += nthr) op[v] = scale * __expf(lg[v] - mx);
}

// ---------- pointer scatter: out[s,b,t, story[b,l]] += (1-pgen)*prob[s,b,l] ----------
__global__ void scatter_kernel(const float* __restrict__ prob, const float* __restrict__ pgen,
                               const int* __restrict__ story, float* __restrict__ out, int t) {
  int idx = blockIdx.x * blockDim.x + threadIdx.x;
  if (idx >= S_ * B_ * L_) return;
  int l = idx % L_;
  int b = (idx / L_) % B_;
  int s = idx / (L_ * B_);
  float w = (1.0f - pgen[(size_t)s * B_ + b]) * prob[idx];
  int v = story[b * L_ + l];
  atomicAdd(out + (((size_t)s * B_ + b) * T_ + t) * (size_t)V_ + v, w);
}

// ---------- teacher-forcing next input: dec_in = embedding[target[b,s,t]] ----------
__global__ void next_input_kernel(const int* __restrict__ tgt, const float* __restrict__ emb,
                                  float* __restrict__ dinNext, int t) {
  int idx = blockIdx.x * blockDim.x + threadIdx.x;
  if (idx >= S_ * B_ * H_) return;
  int j = idx % H_;
  int sb = idx / H_;
  int b = sb % B_;
  int s = sb / B_;
  int tok = tgt[(b * S_ + s) * T_ + t];
  dinNext[idx] = emb[(size_t)tok * H_ + j];
}

// ---------- init: h = encoded_hidden (bcast over s), dec_in = dom_emb + val_emb ----------
__global__ void init_kernel(const float* __restrict__ encH, const float* __restrict__ slotTab,
                            const int* __restrict__ dom, const int* __restrict__ val,
                            float* __restrict__ h0, float* __restrict__ din0) {
  int idx = blockIdx.x * blockDim.x + threadIdx.x;
  if (idx >= S_ * B_ * H_) return;
  int j = idx % H_;
  int sb = idx / H_;
  int b = sb % B_;
  int s = sb / B_;
  h0[idx] = encH[b * H_ + j];
  din0[idx] = slotTab[(size_t)dom[s] * H_ + j] + slotTab[(size_t)val[s] * H_ + j];
}

extern "C" void kernel_launch(void* const* d_in, const int* in_sizes, int n_in,
                              void* d_out, int out_size, void* d_ws, size_t ws_size,
                              hipStream_t stream) {
  const float* encH    = (const float*)d_in[0];
  const float* enc     = (const float*)d_in[1];
  const int*   lens    = (const int*)d_in[2];
  const int*   story   = (const int*)d_in[3];
  const int*   tgt     = (const int*)d_in[4];
  const float* emb     = (const float*)d_in[5];
  const float* w_ih    = (const float*)d_in[6];
  const float* w_hh    = (const float*)d_in[7];
  const float* b_ih    = (const float*)d_in[8];
  const float* b_hh    = (const float*)d_in[9];
  const float* Wr      = (const float*)d_in[10];
  const float* br      = (const float*)d_in[11];
  const float* Wg      = (const float*)d_in[12];
  const float* bg      = (const float*)d_in[13];
  const float* slotTab = (const float*)d_in[14];
  const int*   dom     = (const int*)d_in[15];
  const int*   val     = (const int*)d_in[16];

  float* out   = (float*)d_out;                       // point [S,B,T,V]
  float* gates = out + (size_t)S_ * B_ * T_ * V_;     // gates [S,B,G]

  // workspace carve-up
  float* ws = (float*)d_ws;
  size_t off = 0;
  const size_t SBH = (size_t)S_ * B_ * H_;
  float* hA     = ws + off; off += SBH;
  float* hB     = ws + off; off += SBH;
  float* dinA   = ws + off; off += SBH;
  float* dinB   = ws + off; off += SBH;
  float* gi     = ws + off; off += (size_t)S_ * B_ * 3 * H_;
  float* gh     = ws + off; off += (size_t)S_ * B_ * 3 * H_;
  float* prob   = ws + off; off += (size_t)S_ * B_ * L_;
  float* pgen   = ws + off; off += (size_t)S_ * B_;
  float* logits = ws + off; off += (size_t)S_ * B_ * V_;

  const int thr = 256;
  const int nSBH = (int)((SBH + thr - 1) / thr);

  init_kernel<<<nSBH, thr, 0, stream>>>(encH, slotTab, dom, val, hA, dinA);

  float *hPrev = hA, *hCur = hB, *dinCur = dinA, *dinNext = dinB;

  for (int t = 0; t < T_; ++t) {
    // GRU gate GEMMs: N = 3H = 1200 -> 3 blocks of 400 cols
    dim3 gGru(3 * H_ / BLK_COLS, S_);
    wmma_gemm_lds_kernel<<<gGru, WV * 32, 0, stream>>>(dinCur, w_ih, b_ih, gi, 3 * H_);
    wmma_gemm_lds_kernel<<<gGru, WV * 32, 0, stream>>>(hPrev, w_hh, b_hh, gh, 3 * H_);
    gru_elem_kernel<<<nSBH, thr, 0, stream>>>(gi, gh, hPrev, hCur);

    // attention + context + pgen (+ gates at t==0)
    attn_kernel<<<dim3(B_, S_), 256, 0, stream>>>(hCur, dinCur, enc, lens, Wg, bg, Wr, br,
                                                  prob, pgen, gates, t);

    // vocab logits GEMM: N = V = 18000 -> 45 blocks of 400 cols
    wmma_gemm_lds_kernel<<<dim3(V_ / BLK_COLS, S_), WV * 32, 0, stream>>>(hCur, emb, nullptr,
                                                                          logits, V_);

    // softmax over V, write pgen * p_vocab
    vocab_softmax_kernel<<<dim3(B_, S_), 256, 0, stream>>>(logits, pgen, out, t);

    // pointer distribution scatter-add
    scatter_kernel<<<(S_ * B_ * L_ + thr - 1) / thr, thr, 0, stream>>>(prob, pgen, story, out, t);

    // next teacher-forcing input
    if (t + 1 < T_)
      next_input_kernel<<<nSBH, thr, 0, stream>>>(tgt, emb, dinNext, t);

    float* tmp;
    tmp = hPrev; hPrev = hCur; hCur = tmp;
    tmp = dinCur; dinCur = dinNext; dinNext = tmp;
  }
}